// GTLayer_11905649344581
// MI455X (gfx1250) — compile-verified
//
#include <hip/hip_runtime.h>

typedef __attribute__((ext_vector_type(2))) float v2f;
typedef __attribute__((ext_vector_type(8))) float v8f;

#define NNODES   4096
#define NTYPES   5
#define NCH      2
#define NEDGES   200000

// --------------------------------------------------------------------------
// softmax of conv weights [2,5]; writes f1|f2 (20 floats) to ws scratch and
// to the W1|W2 tail of d_out.
// --------------------------------------------------------------------------
__global__ void gt_softmax_kernel(const float* __restrict__ w1,
                                  const float* __restrict__ w2,
                                  float* __restrict__ fs,
                                  float* __restrict__ wout) {
  if (threadIdx.x == 0 && blockIdx.x == 0) {
    for (int s = 0; s < 2; ++s) {
      const float* w = s ? w2 : w1;
      for (int c = 0; c < NCH; ++c) {
        float mx = w[c * NTYPES];
        for (int k = 1; k < NTYPES; ++k) mx = fmaxf(mx, w[c * NTYPES + k]);
        float e[NTYPES];
        float sum = 0.f;
        for (int k = 0; k < NTYPES; ++k) {
          e[k] = expf(w[c * NTYPES + k] - mx);
          sum += e[k];
        }
        float inv = 1.0f / sum;
        for (int k = 0; k < NTYPES; ++k) {
          float f = e[k] * inv;
          fs[s * 10 + c * NTYPES + k]   = f;
          wout[s * 10 + c * NTYPES + k] = f;
        }
      }
    }
  }
}

// --------------------------------------------------------------------------
// zero the A|B workspace (must happen every call: atomics accumulate)
// --------------------------------------------------------------------------
__global__ __launch_bounds__(256) void gt_zero_kernel(float4* __restrict__ p,
                                                      long n4) {
  long i = (long)blockIdx.x * blockDim.x + threadIdx.x;
  long stride = (long)gridDim.x * blockDim.x;
  float4 z; z.x = 0.f; z.y = 0.f; z.z = 0.f; z.w = 0.f;
  for (; i < n4; i += stride) p[i] = z;
}

// --------------------------------------------------------------------------
// scatter edges, folding the softmax mix in: 4 fp32 atomic adds per edge.
// blockIdx.y = edge type -> mix weights are block-uniform (scalar loads).
// --------------------------------------------------------------------------
__device__ __forceinline__ void atomic_add_f32(float* p, float v) {
  __hip_atomic_fetch_add(p, v, __ATOMIC_RELAXED, __HIP_MEMORY_SCOPE_AGENT);
}

__global__ __launch_bounds__(256) void gt_scatter_kernel(
    const int* __restrict__ ei, const float* __restrict__ ev,
    const float* __restrict__ fs, float* __restrict__ AB) {
  const int k = blockIdx.y;
  const int e = blockIdx.x * blockDim.x + threadIdx.x;
  if (e >= NEDGES) return;
  const float a0 = fs[k];
  const float a1 = fs[NTYPES + k];
  const float b0 = fs[2 * NTYPES + k];
  const float b1 = fs[3 * NTYPES + k];
  const int r = ei[(size_t)k * 2 * NEDGES + e];
  const int c = ei[(size_t)k * 2 * NEDGES + NEDGES + e];
  const float v = ev[(size_t)k * NEDGES + e];
  const size_t NN = (size_t)NNODES * NNODES;
  const size_t off = (size_t)r * NNODES + c;
  atomic_add_f32(&AB[off],          a0 * v);
  atomic_add_f32(&AB[NN + off],     a1 * v);
  atomic_add_f32(&AB[2 * NN + off], b0 * v);
  atomic_add_f32(&AB[3 * NN + off], b1 * v);
}

// --------------------------------------------------------------------------
// fp32 WMMA GEMM: H[c] = A[c] @ B[c], 4096x4096x4096 per channel.
// Block: 256 threads = 8 wave32s, tile 128x128, K-chunk 16.
// Wave grid 4x2 -> each wave computes 32x64 = 2x4 accumulators of 16x16.
// LDS row stride 20 floats -> conflict-free ds_load_b64 fragment loads.
// --------------------------------------------------------------------------
__global__ __launch_bounds__(256) void gt_gemm_wmma_kernel(
    const float* __restrict__ Amat, const float* __restrict__ Bmat,
    float* __restrict__ H) {
  constexpr int N  = NNODES;
  constexpr int LD = 20;
  __shared__ __align__(16) float sA[128 * LD];   // A tile, row-major [m][k]
  __shared__ __align__(16) float sBT[128 * LD];  // B tile, transposed [n][k]

  const size_t NN = (size_t)N * N;
  const int ch = blockIdx.z;
  const float* __restrict__ Ab = Amat + (size_t)ch * NN;
  const float* __restrict__ Bb = Bmat + (size_t)ch * NN;
  float* __restrict__ Hb = H + (size_t)ch * NN;

  const int rowBase = blockIdx.y * 128;
  const int colBase = blockIdx.x * 128;
  const int tid  = threadIdx.x;
  const int lane = tid & 31;
  const int wid  = tid >> 5;
  const int mWave = (wid >> 1) * 32;  // 0,32,64,96
  const int nWave = (wid & 1) * 64;   // 0,64
  const int half  = lane >> 4;        // K-pair selector per WMMA layout
  const int l16   = lane & 15;        // M (A frag) / N (B frag) index

  v8f acc[2][4];
  for (int mi = 0; mi < 2; ++mi)
    for (int ni = 0; ni < 4; ++ni)
      acc[mi][ni] = (v8f){0.f, 0.f, 0.f, 0.f, 0.f, 0.f, 0.f, 0.f};

  // A staging: thread -> rows (tid/4, tid/4+64), float4 at k-offset (tid%4)*4
  const int rA  = tid >> 2;
  const int kvA = (tid & 3) << 2;
  // B staging: thread -> column tid%128, eight k's starting at (tid/128)*8
  const int nB = tid & 127;
  const int kB = (tid >> 7) * 8;

  for (int kt = 0; kt < N / 16; ++kt) {
    const int kBase = kt * 16;

    const float4 a0 = *(const float4*)&Ab[(size_t)(rowBase + rA) * N + kBase + kvA];
    const float4 a1 = *(const float4*)&Ab[(size_t)(rowBase + rA + 64) * N + kBase + kvA];
    float bv[8];
#pragma unroll
    for (int i = 0; i < 8; ++i)
      bv[i] = Bb[(size_t)(kBase + kB + i) * N + colBase + nB];

    __syncthreads();  // previous chunk's compute done before overwriting LDS
    *(float4*)&sA[rA * LD + kvA]        = a0;
    *(float4*)&sA[(rA + 64) * LD + kvA] = a1;
#pragma unroll
    for (int i = 0; i < 8; ++i)
      sBT[nB * LD + kB + i] = bv[i];
    __syncthreads();

#pragma unroll
    for (int kk = 0; kk < 4; ++kk) {
      const int ko = kk * 4 + half * 2;
      v2f af[2], bf[4];
#pragma unroll
      for (int mi = 0; mi < 2; ++mi)
        af[mi] = *(const v2f*)&sA[(mWave + mi * 16 + l16) * LD + ko];
#pragma unroll
      for (int ni = 0; ni < 4; ++ni)
        bf[ni] = *(const v2f*)&sBT[(nWave + ni * 16 + l16) * LD + ko];
#pragma unroll
      for (int mi = 0; mi < 2; ++mi)
#pragma unroll
        for (int ni = 0; ni < 4; ++ni)
          acc[mi][ni] = __builtin_amdgcn_wmma_f32_16x16x4_f32(
              false, af[mi], false, bf[ni], (short)0, acc[mi][ni],
              false, false);
    }
  }

  // epilogue: C/D layout -> VGPR j holds row (j + 8*half), col = l16
#pragma unroll
  for (int mi = 0; mi < 2; ++mi)
#pragma unroll
    for (int ni = 0; ni < 4; ++ni) {
      const int row0 = rowBase + mWave + mi * 16 + half * 8;
      const int col  = colBase + nWave + ni * 16 + l16;
#pragma unroll
      for (int j = 0; j < 8; ++j)
        Hb[(size_t)(row0 + j) * N + col] = acc[mi][ni][j];
    }
}

// --------------------------------------------------------------------------
extern "C" void kernel_launch(void* const* d_in, const int* in_sizes, int n_in,
                              void* d_out, int out_size, void* d_ws, size_t ws_size,
                              hipStream_t stream) {
  const int*   ei = (const int*)d_in[0];    // [5,2,200000]
  const float* ev = (const float*)d_in[1];  // [5,200000]
  const float* w1 = (const float*)d_in[2];  // [2,5]
  const float* w2 = (const float*)d_in[3];  // [2,5]
  float* out = (float*)d_out;               // H [2,4096,4096] | W1[10] | W2[10]

  const size_t NN = (size_t)NNODES * NNODES;
  float* AB = (float*)d_ws;        // A(c0),A(c1),B(c0),B(c1) : 4*NN floats
  float* fs = AB + 4 * NN;         // f1|f2 mix weights : 20 floats
  float* Hout = out;
  float* Wout = out + (size_t)NCH * NN;

  gt_softmax_kernel<<<1, 32, 0, stream>>>(w1, w2, fs, Wout);

  const long n4 = (long)(4 * NN / 4);
  gt_zero_kernel<<<4096, 256, 0, stream>>>((float4*)AB, n4);

  dim3 sg((NEDGES + 255) / 256, NTYPES);
  gt_scatter_kernel<<<sg, 256, 0, stream>>>(ei, ev, fs, AB);

  dim3 gg(NNODES / 128, NNODES / 128, NCH);
  gt_gemm_wmma_kernel<<<gg, 256, 0, stream>>>(AB, AB + 2 * NN, Hout);
}